// PaiNNInteraction_17514876634209
// MI455X (gfx1250) — compile-verified
//
#include <hip/hip_runtime.h>

typedef float v2f __attribute__((ext_vector_type(2)));
typedef float v8f __attribute__((ext_vector_type(8)));

#define NN 20000   // nodes
#define NE 160000  // edges
#define HH 128     // hidden
#define H3 384     // 3*H
#define RBF 20

__device__ __forceinline__ float silu_f(float v) { return v / (1.0f + __expf(-v)); }

// D = A(16x4 f32) * B(4x16 f32) + C(16x16 f32), full fp32 WMMA.
__device__ __forceinline__ v8f wmma_f32(v2f a, v2f b, v8f c) {
    return __builtin_amdgcn_wmma_f32_16x16x4_f32(false, a, false, b, (short)0, c, false, false);
}

// ---------------------------------------------------------------------------
// Node MLP: x = silu(q @ W1 + b1) @ W2 + b2        [N, 3H]
// One workgroup (8 wave32) per 16-node row tile. Each wave owns 3 column
// tiles with 3 live accumulators -> 3 independent WMMA chains per K-step.
// ---------------------------------------------------------------------------
__global__ __launch_bounds__(256) void node_mlp_kernel(
    const float* __restrict__ q,
    const float* __restrict__ W1, const float* __restrict__ b1,
    const float* __restrict__ W2, const float* __restrict__ b2,
    float* __restrict__ x_out)
{
    __shared__ float h1[16 * 388];                // 16 x 384 (+pad)
    const int m0   = blockIdx.x * 16;
    const int wave = threadIdx.x >> 5;
    const int lane = threadIdx.x & 31;
    const int lo   = lane & 15;
    const int hi   = lane >> 4;                   // 0/1 -> K offset 0/2, row offset 0/8

    const int n0 = (wave + 0)  * 16 + lo;         // column owned in tile 0
    const int n1 = (wave + 8)  * 16 + lo;         // column owned in tile 1
    const int n2 = (wave + 16) * 16 + lo;         // column owned in tile 2

    // ---- stage 1: h1 = silu(q[16,128] @ W1[128,384] + b1) ----
    {
        v8f acc0, acc1, acc2;
        const float bi0 = b1[n0], bi1 = b1[n1], bi2 = b1[n2];
        #pragma unroll
        for (int v = 0; v < 8; ++v) { acc0[v] = bi0; acc1[v] = bi1; acc2[v] = bi2; }
        const float* qrow = q + (size_t)(m0 + lo) * HH + 2 * hi;
        for (int kk = 0; kk < HH; kk += 4) {
            v2f a = *(const v2f*)(qrow + kk);
            const int k0 = kk + 2 * hi;
            v2f b0, b1v, b2v;
            b0.x  = W1[k0 * H3 + n0];       b0.y  = W1[(k0 + 1) * H3 + n0];
            b1v.x = W1[k0 * H3 + n1];       b1v.y = W1[(k0 + 1) * H3 + n1];
            b2v.x = W1[k0 * H3 + n2];       b2v.y = W1[(k0 + 1) * H3 + n2];
            acc0 = wmma_f32(a, b0,  acc0);
            acc1 = wmma_f32(a, b1v, acc1);
            acc2 = wmma_f32(a, b2v, acc2);
        }
        #pragma unroll
        for (int v = 0; v < 8; ++v) {
            const int r = (v + 8 * hi) * 388;
            h1[r + n0] = silu_f(acc0[v]);
            h1[r + n1] = silu_f(acc1[v]);
            h1[r + n2] = silu_f(acc2[v]);
        }
    }
    __syncthreads();

    // ---- stage 2: x = h1[16,384] @ W2[384,384] + b2 ----
    {
        v8f acc0, acc1, acc2;
        const float bi0 = b2[n0], bi1 = b2[n1], bi2 = b2[n2];
        #pragma unroll
        for (int v = 0; v < 8; ++v) { acc0[v] = bi0; acc1[v] = bi1; acc2[v] = bi2; }
        const float* hrow = &h1[lo * 388 + 2 * hi];
        for (int kk = 0; kk < H3; kk += 4) {
            v2f a = *(const v2f*)(hrow + kk);
            const int k0 = kk + 2 * hi;
            v2f b0, b1v, b2v;
            b0.x  = W2[k0 * H3 + n0];       b0.y  = W2[(k0 + 1) * H3 + n0];
            b1v.x = W2[k0 * H3 + n1];       b1v.y = W2[(k0 + 1) * H3 + n1];
            b2v.x = W2[k0 * H3 + n2];       b2v.y = W2[(k0 + 1) * H3 + n2];
            acc0 = wmma_f32(a, b0,  acc0);
            acc1 = wmma_f32(a, b1v, acc1);
            acc2 = wmma_f32(a, b2v, acc2);
        }
        #pragma unroll
        for (int v = 0; v < 8; ++v) {
            float* orow = x_out + (size_t)(m0 + v + 8 * hi) * H3;
            orow[n0] = acc0[v];
            orow[n1] = acc1[v];
            orow[n2] = acc2[v];
        }
    }
}

// ---------------------------------------------------------------------------
// Edge kernel: filter network (2 fused WMMA GEMMs, filter tile kept in LDS)
// followed by the gather / message / atomic-scatter phase for 16 edges.
// ---------------------------------------------------------------------------
__global__ __launch_bounds__(256) void edge_kernel(
    const float* __restrict__ rbf,
    const float* __restrict__ Wf1, const float* __restrict__ bf1,
    const float* __restrict__ Wf2, const float* __restrict__ bf2,
    const float* __restrict__ cutoff,
    const int*   __restrict__ edge_index,
    const float* __restrict__ x,
    const float* __restrict__ mu,
    const float* __restrict__ uv,
    float* __restrict__ sAcc, float* __restrict__ vAcc, float* __restrict__ deg)
{
    __shared__ float f1s[16 * 132];               // 16 x 128 (+pad)
    __shared__ float filt[16 * 388];              // 16 x 384 (+pad)
    __shared__ float cut[16];
    __shared__ int   sidx[16], tidx[16];

    const int e0   = blockIdx.x * 16;
    const int wave = threadIdx.x >> 5;
    const int lane = threadIdx.x & 31;
    const int lo   = lane & 15;
    const int hi   = lane >> 4;

    if (threadIdx.x < 16) {
        tidx[threadIdx.x] = edge_index[e0 + threadIdx.x];        // row 0 = target
        sidx[threadIdx.x] = edge_index[NE + e0 + threadIdx.x];   // row 1 = source
        cut[threadIdx.x]  = cutoff[e0 + threadIdx.x];
    }

    // ---- stage 1: f1 = silu(rbf[16,20] @ Wf1[20,128] + bf1), one tile/wave ----
    {
        const int n = wave * 16 + lo;
        const float bias = bf1[n];
        v8f acc;
        #pragma unroll
        for (int v = 0; v < 8; ++v) acc[v] = bias;
        #pragma unroll
        for (int kk = 0; kk < RBF; kk += 4) {     // K = 20 -> 5 steps
            v2f a = *(const v2f*)(rbf + (size_t)(e0 + lo) * RBF + kk + 2 * hi);
            v2f b;
            b.x = Wf1[(kk + 2 * hi)     * HH + n];
            b.y = Wf1[(kk + 2 * hi + 1) * HH + n];
            acc = wmma_f32(a, b, acc);
        }
        #pragma unroll
        for (int v = 0; v < 8; ++v)
            f1s[(v + 8 * hi) * 132 + n] = silu_f(acc[v]);
    }
    __syncthreads();

    // ---- stage 2: filters = (f1 @ Wf2[128,384] + bf2) * cutoff ----
    {
        const int n0 = (wave + 0)  * 16 + lo;
        const int n1 = (wave + 8)  * 16 + lo;
        const int n2 = (wave + 16) * 16 + lo;
        v8f acc0, acc1, acc2;
        const float bi0 = bf2[n0], bi1 = bf2[n1], bi2 = bf2[n2];
        #pragma unroll
        for (int v = 0; v < 8; ++v) { acc0[v] = bi0; acc1[v] = bi1; acc2[v] = bi2; }
        const float* frow = &f1s[lo * 132 + 2 * hi];
        for (int kk = 0; kk < HH; kk += 4) {
            v2f a = *(const v2f*)(frow + kk);
            const int k0 = kk + 2 * hi;
            v2f b0, b1v, b2v;
            b0.x  = Wf2[k0 * H3 + n0];      b0.y  = Wf2[(k0 + 1) * H3 + n0];
            b1v.x = Wf2[k0 * H3 + n1];      b1v.y = Wf2[(k0 + 1) * H3 + n1];
            b2v.x = Wf2[k0 * H3 + n2];      b2v.y = Wf2[(k0 + 1) * H3 + n2];
            acc0 = wmma_f32(a, b0,  acc0);
            acc1 = wmma_f32(a, b1v, acc1);
            acc2 = wmma_f32(a, b2v, acc2);
        }
        #pragma unroll
        for (int v = 0; v < 8; ++v) {
            const int r = v + 8 * hi;             // edge row within tile
            const float c = cut[r];
            filt[r * 388 + n0] = acc0[v] * c;
            filt[r * 388 + n1] = acc1[v] * c;
            filt[r * 388 + n2] = acc2[v] * c;
        }
    }
    __syncthreads();

    // ---- stage 3: edge messages + scatter (128 threads per channel dim) ----
    const int h = threadIdx.x & 127;
    for (int el = (threadIdx.x >> 7); el < 16; el += 2) {
        const int e = e0 + el;
        const int s = sidx[el];
        const int t = tidx[el];
        const float fq = filt[el * 388 + h];
        const float fr = filt[el * 388 + HH + h];
        const float fm = filt[el * 388 + 2 * HH + h];
        const float* xs = x + (size_t)s * H3;
        const float msq = xs[h]          * fq;    // scalar message
        const float msr = xs[HH + h]     * fr;    // radial gate
        const float msm = xs[2 * HH + h] * fm;    // mu gate
        atomicAdd(&sAcc[(size_t)t * HH + h], msq);
        const float u0 = uv[e * 3 + 0];
        const float u1 = uv[e * 3 + 1];
        const float u2 = uv[e * 3 + 2];
        const float* ms = mu + (size_t)s * H3;
        atomicAdd(&vAcc[((size_t)t * 3 + 0) * HH + h], u0 * msr + ms[h]          * msm);
        atomicAdd(&vAcc[((size_t)t * 3 + 1) * HH + h], u1 * msr + ms[HH + h]     * msm);
        atomicAdd(&vAcc[((size_t)t * 3 + 2) * HH + h], u2 * msr + ms[2 * HH + h] * msm);
    }
    if (threadIdx.x < 16) atomicAdd(&deg[tidx[threadIdx.x]], 1.0f);
}

// ---------------------------------------------------------------------------
// Finalize: q_out = q + s/deg ; mu_out = mu + v/deg  (deg clamped to >= 1)
// ---------------------------------------------------------------------------
__global__ __launch_bounds__(256) void finalize_kernel(
    const float* __restrict__ q, const float* __restrict__ mu,
    const float* __restrict__ sAcc, const float* __restrict__ vAcc,
    const float* __restrict__ deg, float* __restrict__ out)
{
    const int i = blockIdx.x * 256 + threadIdx.x;    // over N * 512
    if (i >= NN * 512) return;
    const int n = i >> 9;
    const int j = i & 511;
    const float d = fmaxf(deg[n], 1.0f);
    if (j < HH) {
        out[(size_t)n * HH + j] = q[(size_t)n * HH + j] + sAcc[(size_t)n * HH + j] / d;
    } else {
        const int k = j - HH;
        out[(size_t)NN * HH + (size_t)n * H3 + k] =
            mu[(size_t)n * H3 + k] + vAcc[(size_t)n * H3 + k] / d;
    }
}

extern "C" void kernel_launch(void* const* d_in, const int* in_sizes, int n_in,
                              void* d_out, int out_size, void* d_ws, size_t ws_size,
                              hipStream_t stream) {
    (void)in_sizes; (void)n_in; (void)out_size; (void)ws_size;
    const float* q      = (const float*)d_in[0];
    const float* mu     = (const float*)d_in[1];
    const int*   ei     = (const int*)  d_in[2];
    const float* rbf    = (const float*)d_in[3];
    const float* uv     = (const float*)d_in[4];
    const float* cut    = (const float*)d_in[5];
    const float* W1     = (const float*)d_in[6];
    const float* b1     = (const float*)d_in[7];
    const float* W2     = (const float*)d_in[8];
    const float* b2     = (const float*)d_in[9];
    const float* Wf1    = (const float*)d_in[10];
    const float* bf1    = (const float*)d_in[11];
    const float* Wf2    = (const float*)d_in[12];
    const float* bf2    = (const float*)d_in[13];

    float* ws = (float*)d_ws;
    float* xbuf = ws;                               // [N, 3H]
    float* sA   = xbuf + (size_t)NN * H3;           // [N, H]
    float* vA   = sA   + (size_t)NN * HH;           // [N, 3, H]
    float* dg   = vA   + (size_t)NN * H3;           // [N]

    // zero scatter accumulators (graph-capture-safe memset node)
    hipMemsetAsync(sA, 0, (size_t)NN * (HH + H3 + 1) * sizeof(float), stream);

    node_mlp_kernel<<<NN / 16, 256, 0, stream>>>(q, W1, b1, W2, b2, xbuf);
    edge_kernel<<<NE / 16, 256, 0, stream>>>(rbf, Wf1, bf1, Wf2, bf2, cut, ei,
                                             xbuf, mu, uv, sA, vA, dg);
    finalize_kernel<<<(NN * 512 + 255) / 256, 256, 0, stream>>>(q, mu, sA, vA, dg,
                                                                (float*)d_out);
}